// EncoderLayer_32478542693006
// MI455X (gfx1250) — compile-verified
//
#include <hip/hip_runtime.h>
#include <math.h>

// ---------------------------------------------------------------------------
// CDNA5 (gfx1250) wave32 WMMA fp32 path: V_WMMA_F32_16X16X4_F32
// A: 16x4 f32 (2 VGPRs/lane), B: 4x16 f32 (2 VGPRs/lane), C/D: 16x16 f32 (8 VGPRs)
// ---------------------------------------------------------------------------
typedef __attribute__((ext_vector_type(2))) float v2f;
typedef __attribute__((ext_vector_type(8))) float v8f;

#define WMMA_F32_16x16x4(a, b, c) \
  __builtin_amdgcn_wmma_f32_16x16x4_f32(false, (a), false, (b), (short)0, (c), false, false)

__device__ __forceinline__ float lrelu(float v) { return v >= 0.f ? v : 0.2f * v; }

__device__ __forceinline__ void atomicMaxF(float* addr, float v) {
  // monotonic int mapping: works with -inf init
  if (v >= 0.f) atomicMax((int*)addr, __float_as_int(v));
  else          atomicMin((unsigned int*)addr, __float_as_uint(v));
}

// ---------------------------------------------------------------------------
// Kernel 0: init workspace (mx = -inf, den = 0, agg = 0) every call
// ---------------------------------------------------------------------------
__global__ void k_init(float* __restrict__ mx, float* __restrict__ den,
                       float* __restrict__ agg, int N) {
  size_t i = (size_t)blockIdx.x * blockDim.x + threadIdx.x;
  size_t tot = (size_t)N * 512;
  for (size_t j = i; j < tot; j += (size_t)gridDim.x * blockDim.x) agg[j] = 0.f;
  if (i < (size_t)N * 4) { mx[i] = -INFINITY; den[i] = 0.f; }
}

// ---------------------------------------------------------------------------
// Kernel 1: LN1 + xs = y@Wl+bl, xd = y@Wr+br   (N x 128) x (128 x 512) twice
// block = 256 threads = 8 waves, 16 rows per block, WMMA f32 16x16x4
// ---------------------------------------------------------------------------
__global__ __launch_bounds__(256) void k_ln1_xsxd(
    const float* __restrict__ x, const float* __restrict__ g, const float* __restrict__ b,
    const float* __restrict__ Wl, const float* __restrict__ bl,
    const float* __restrict__ Wr, const float* __restrict__ br,
    float* __restrict__ xs, float* __restrict__ xd, int N) {
  __shared__ float yln[16][132];  // pad 132 -> conflict-free column reads
  const int tid = threadIdx.x, lane = tid & 31, wid = tid >> 5;
  const int rowBase = blockIdx.x * 16;

  // --- LayerNorm: wave wid normalizes rows 2*wid and 2*wid+1 (half-wave each)
  {
    const int r = 2 * wid + (lane >> 4);
    const int hl = lane & 15;
    const int row = rowBase + r;
    float vals[8];
    float s = 0.f, sq = 0.f;
    if (row < N) {
      const float* xr = x + (size_t)row * 128 + hl * 8;
#pragma unroll
      for (int i = 0; i < 8; ++i) { float v = xr[i]; vals[i] = v; s += v; sq += v * v; }
    } else {
#pragma unroll
      for (int i = 0; i < 8; ++i) vals[i] = 0.f;
    }
#pragma unroll
    for (int off = 8; off >= 1; off >>= 1) {
      s += __shfl_xor(s, off, 16);
      sq += __shfl_xor(sq, off, 16);
    }
    const float mu = s * (1.f / 128.f);
    const float var = sq * (1.f / 128.f) - mu * mu;
    const float rs = rsqrtf(var + 1e-5f);
#pragma unroll
    for (int i = 0; i < 8; ++i) {
      const int c = hl * 8 + i;
      yln[r][c] = (vals[i] - mu) * rs * g[c] + b[c];
    }
  }
  __syncthreads();

  // --- WMMA: 64 output tiles of 16x16 across combined 1024 columns (Wl|Wr)
  const int m = lane & 15;           // A row
  const int n = lane & 15;           // B/D col
  const int kb = (lane >> 4) * 2;    // K sub-offset per half-wave
  for (int t = wid; t < 64; t += 8) {
    const float* W; const float* bias; float* out; int colLocal;
    if (t < 32) { W = Wl; bias = bl; out = xs; colLocal = t * 16; }
    else        { W = Wr; bias = br; out = xd; colLocal = (t - 32) * 16; }
    v8f acc = {};
#pragma unroll
    for (int k0 = 0; k0 < 128; k0 += 4) {
      v2f a, bf;
      a[0] = yln[m][k0 + kb];
      a[1] = yln[m][k0 + kb + 1];
      bf[0] = W[(size_t)(k0 + kb) * 512 + colLocal + n];
      bf[1] = W[(size_t)(k0 + kb + 1) * 512 + colLocal + n];
      acc = WMMA_F32_16x16x4(a, bf, acc);
    }
    const float bv = bias[colLocal + n];
#pragma unroll
    for (int r = 0; r < 8; ++r) {
      const int mr = (lane < 16) ? r : r + 8;
      const int row = rowBase + mr;
      if (row < N) out[(size_t)row * 512 + colLocal + n] = acc[r] + bv;
    }
  }
}

// ---------------------------------------------------------------------------
// Kernel 2: per-edge attention logits + segment max (wave per edge)
// logits[e,h] = sum_c lrelu(xs[src,h,c]+xd[dst,h,c]+w*We[h,c]) * att[h,c]
// ---------------------------------------------------------------------------
__global__ __launch_bounds__(256) void k_edge_logits(
    const int* __restrict__ ei, const float* __restrict__ ew,
    const float* __restrict__ xs, const float* __restrict__ xd,
    const float* __restrict__ We, const float* __restrict__ att,
    float* __restrict__ logits, float* __restrict__ mx, int E) {
  __shared__ float sWe[512], sAtt[512];
  const int tid = threadIdx.x;
  sWe[tid] = We[tid];       sWe[tid + 256] = We[tid + 256];
  sAtt[tid] = att[tid];     sAtt[tid + 256] = att[tid + 256];
  __syncthreads();

  const int lane = tid & 31, wid = tid >> 5;
  const int e = blockIdx.x * 8 + wid;
  if (e >= E) return;
  const int s = ei[e];
  const int d = ei[E + e];
  const float w = ew[e];
  const float4* xs4 = (const float4*)(xs + (size_t)s * 512);
  const float4* xd4 = (const float4*)(xd + (size_t)d * 512);
  const float4* We4 = (const float4*)sWe;
  const float4* at4 = (const float4*)sAtt;

#pragma unroll
  for (int h = 0; h < 4; ++h) {
    const int idx = h * 32 + lane;
    float4 a = xs4[idx];
    float4 bb = xd4[idx];
    float4 we = We4[idx];
    float4 at = at4[idx];
    float acc = lrelu(a.x + bb.x + w * we.x) * at.x
              + lrelu(a.y + bb.y + w * we.y) * at.y
              + lrelu(a.z + bb.z + w * we.z) * at.z
              + lrelu(a.w + bb.w + w * we.w) * at.w;
#pragma unroll
    for (int off = 16; off >= 1; off >>= 1) acc += __shfl_xor(acc, off, 32);
    if (lane == 0) {
      logits[(size_t)e * 4 + h] = acc;
      atomicMaxF(&mx[(size_t)d * 4 + h], acc);
    }
  }
}

// ---------------------------------------------------------------------------
// Kernel 3: ex = exp(logit - mx[dst]); den[dst] += ex  (thread per (e,h))
// ---------------------------------------------------------------------------
__global__ void k_softmax_den(const int* __restrict__ ei, float* __restrict__ logits,
                              const float* __restrict__ mx, float* __restrict__ den, int E) {
  const int i = blockIdx.x * blockDim.x + threadIdx.x;
  if (i >= E * 4) return;
  const int e = i >> 2, h = i & 3;
  const int d = ei[E + e];
  const float v = expf(logits[i] - mx[(size_t)d * 4 + h]);
  logits[i] = v;
  atomicAdd(&den[(size_t)d * 4 + h], v);
}

// ---------------------------------------------------------------------------
// Kernel 4: agg[dst] += xs[src] * alpha  (wave per edge, f32 atomics -> L2)
// ---------------------------------------------------------------------------
__global__ __launch_bounds__(256) void k_scatter(
    const int* __restrict__ ei, const float* __restrict__ xs,
    const float* __restrict__ logits, const float* __restrict__ den,
    float* __restrict__ agg, int E) {
  const int lane = threadIdx.x & 31, wid = threadIdx.x >> 5;
  const int e = blockIdx.x * 8 + wid;
  if (e >= E) return;
  const int s = ei[e];
  const int d = ei[E + e];
  const float4* xs4 = (const float4*)(xs + (size_t)s * 512);
  float* aggd = agg + (size_t)d * 512;
#pragma unroll
  for (int h = 0; h < 4; ++h) {
    const float alpha = logits[(size_t)e * 4 + h] / (den[(size_t)d * 4 + h] + 1e-16f);
    float4 v = xs4[h * 32 + lane];
    const int c = h * 128 + lane * 4;
    atomicAdd(&aggd[c + 0], v.x * alpha);
    atomicAdd(&aggd[c + 1], v.y * alpha);
    atomicAdd(&aggd[c + 2], v.z * alpha);
    atomicAdd(&aggd[c + 3], v.w * alpha);
  }
}

// ---------------------------------------------------------------------------
// Kernel 5: xnew = x + mean_h(agg) + bias_att ; y2 = LN2(xnew)  (wave per node)
// ---------------------------------------------------------------------------
__global__ __launch_bounds__(256) void k_residual_ln2(
    const float* __restrict__ x, const float* __restrict__ agg,
    const float* __restrict__ bias_att, const float* __restrict__ g,
    const float* __restrict__ b, float* __restrict__ xnew, float* __restrict__ y2, int N) {
  const int lane = threadIdx.x & 31, wid = threadIdx.x >> 5;
  const int nid = blockIdx.x * 8 + wid;
  if (nid >= N) return;
  const float4* aggn = (const float4*)(agg + (size_t)nid * 512);
  const float4* xn = (const float4*)(x + (size_t)nid * 128);
  const float4* ba = (const float4*)bias_att;
  float4 a0 = aggn[lane], a1 = aggn[32 + lane], a2 = aggn[64 + lane], a3 = aggn[96 + lane];
  float4 xv = xn[lane];
  float4 bv = ba[lane];
  float4 v;
  v.x = xv.x + 0.25f * (a0.x + a1.x + a2.x + a3.x) + bv.x;
  v.y = xv.y + 0.25f * (a0.y + a1.y + a2.y + a3.y) + bv.y;
  v.z = xv.z + 0.25f * (a0.z + a1.z + a2.z + a3.z) + bv.z;
  v.w = xv.w + 0.25f * (a0.w + a1.w + a2.w + a3.w) + bv.w;
  float s = v.x + v.y + v.z + v.w;
  float sq = v.x * v.x + v.y * v.y + v.z * v.z + v.w * v.w;
#pragma unroll
  for (int off = 16; off >= 1; off >>= 1) {
    s += __shfl_xor(s, off, 32);
    sq += __shfl_xor(sq, off, 32);
  }
  const float mu = s * (1.f / 128.f);
  const float var = sq * (1.f / 128.f) - mu * mu;
  const float rs = rsqrtf(var + 1e-5f);
  ((float4*)(xnew + (size_t)nid * 128))[lane] = v;
  const int c = lane * 4;
  float4 yv;
  yv.x = (v.x - mu) * rs * g[c + 0] + b[c + 0];
  yv.y = (v.y - mu) * rs * g[c + 1] + b[c + 1];
  yv.z = (v.z - mu) * rs * g[c + 2] + b[c + 2];
  yv.w = (v.w - mu) * rs * g[c + 3] + b[c + 3];
  ((float4*)(y2 + (size_t)nid * 128))[lane] = yv;
}

// ---------------------------------------------------------------------------
// Kernel 6: FFN  out = xnew + gelu(y2@W1+b1)@W2 + b2  (fully fused, WMMA f32)
// block = 256 threads = 8 waves; 16 rows; each wave owns one 16x16 tile.
// ---------------------------------------------------------------------------
__global__ __launch_bounds__(256) void k_ffn(
    const float* __restrict__ xnew, const float* __restrict__ y2,
    const float* __restrict__ W1, const float* __restrict__ b1,
    const float* __restrict__ W2, const float* __restrict__ b2,
    float* __restrict__ out, int N) {
  __shared__ float ytile[16][132];
  __shared__ float htile[16][132];
  const int tid = threadIdx.x, lane = tid & 31, wid = tid >> 5;
  const int rowBase = blockIdx.x * 16;

  {  // stage 0: y2 tile -> LDS
    const int r = tid >> 4, c0 = (tid & 15) * 8;
    const int row = rowBase + r;
    const float* src = y2 + (size_t)row * 128 + c0;
#pragma unroll
    for (int i = 0; i < 8; ++i) ytile[r][c0 + i] = (row < N) ? src[i] : 0.f;
  }
  __syncthreads();

  const int m = lane & 15, n = lane & 15, kb = (lane >> 4) * 2;
  const int colLocal = wid * 16;

  {  // stage 1: h = gelu(y2 @ W1 + b1)
    v8f acc = {};
#pragma unroll
    for (int k0 = 0; k0 < 128; k0 += 4) {
      v2f a, bf;
      a[0] = ytile[m][k0 + kb];
      a[1] = ytile[m][k0 + kb + 1];
      bf[0] = W1[(size_t)(k0 + kb) * 128 + colLocal + n];
      bf[1] = W1[(size_t)(k0 + kb + 1) * 128 + colLocal + n];
      acc = WMMA_F32_16x16x4(a, bf, acc);
    }
    const float bv = b1[colLocal + n];
#pragma unroll
    for (int r = 0; r < 8; ++r) {
      const int mr = (lane < 16) ? r : r + 8;
      const float v = acc[r] + bv;
      htile[mr][colLocal + n] = 0.5f * v * (1.f + erff(v * 0.70710678118654752f));
    }
  }
  __syncthreads();

  {  // stage 2: out = xnew + h @ W2 + b2
    v8f acc = {};
#pragma unroll
    for (int k0 = 0; k0 < 128; k0 += 4) {
      v2f a, bf;
      a[0] = htile[m][k0 + kb];
      a[1] = htile[m][k0 + kb + 1];
      bf[0] = W2[(size_t)(k0 + kb) * 128 + colLocal + n];
      bf[1] = W2[(size_t)(k0 + kb + 1) * 128 + colLocal + n];
      acc = WMMA_F32_16x16x4(a, bf, acc);
    }
    const float bv = b2[colLocal + n];
#pragma unroll
    for (int r = 0; r < 8; ++r) {
      const int mr = (lane < 16) ? r : r + 8;
      const int row = rowBase + mr;
      if (row < N)
        out[(size_t)row * 128 + colLocal + n] =
            xnew[(size_t)row * 128 + colLocal + n] + acc[r] + bv;
    }
  }
}

// ---------------------------------------------------------------------------
extern "C" void kernel_launch(void* const* d_in, const int* in_sizes, int n_in,
                              void* d_out, int out_size, void* d_ws, size_t ws_size,
                              hipStream_t stream) {
  const float* x        = (const float*)d_in[0];
  const int*   ei       = (const int*)  d_in[1];   // [2,E] int32
  const float* ew       = (const float*)d_in[2];   // [E,1]
  const float* ln1_g    = (const float*)d_in[3];
  const float* ln1_b    = (const float*)d_in[4];
  const float* Wl       = (const float*)d_in[5];   // [128,512]
  const float* bl       = (const float*)d_in[6];
  const float* Wr       = (const float*)d_in[7];
  const float* br       = (const float*)d_in[8];
  const float* We       = (const float*)d_in[9];   // [1,512]
  const float* att      = (const float*)d_in[10];  // [4,128]
  const float* bias_att = (const float*)d_in[11];
  const float* ln2_g    = (const float*)d_in[12];
  const float* ln2_b    = (const float*)d_in[13];
  const float* W1       = (const float*)d_in[14];  // [128,128]
  const float* b1       = (const float*)d_in[15];
  const float* W2       = (const float*)d_in[16];  // [128,128]
  const float* b2       = (const float*)d_in[17];

  const int N = in_sizes[0] / 128;
  const int E = in_sizes[2];

  float* ws = (float*)d_ws;
  size_t o = 0;
  float* xs     = ws + o; o += (size_t)N * 512;
  float* xd     = ws + o; o += (size_t)N * 512;
  float* agg    = ws + o; o += (size_t)N * 512;
  float* logits = ws + o; o += (size_t)E * 4;
  float* mx     = ws + o; o += (size_t)N * 4;
  float* den    = ws + o; o += (size_t)N * 4;
  float* xnew   = ws + o; o += (size_t)N * 128;
  float* y2     = ws + o; o += (size_t)N * 128;

  k_init<<<(N * 512 + 255) / 256, 256, 0, stream>>>(mx, den, agg, N);
  k_ln1_xsxd<<<(N + 15) / 16, 256, 0, stream>>>(x, ln1_g, ln1_b, Wl, bl, Wr, br, xs, xd, N);
  k_edge_logits<<<(E + 7) / 8, 256, 0, stream>>>(ei, ew, xs, xd, We, att, logits, mx, E);
  k_softmax_den<<<(E * 4 + 255) / 256, 256, 0, stream>>>(ei, logits, mx, den, E);
  k_scatter<<<(E + 7) / 8, 256, 0, stream>>>(ei, xs, logits, den, agg, E);
  k_residual_ln2<<<(N + 7) / 8, 256, 0, stream>>>(x, agg, bias_att, ln2_g, ln2_b, xnew, y2, N);
  k_ffn<<<(N + 15) / 16, 256, 0, stream>>>(xnew, y2, W1, b1, W2, b2, (float*)d_out, N);
}